// SoundSourceLocalize_10196252361119
// MI455X (gfx1250) — compile-verified
//
#include <hip/hip_runtime.h>
#include <math.h>

// ---------------- problem constants (match the JAX reference) ----------------
#define T_SAMP   6400000
#define NFFT     512
#define HOP      160
#define NBINS    257          // NFFT/2 + 1
#define NFRAMES  39997        // (T - NFFT)/HOP + 1
#define NANGLES  181
#define PRE      0.97f
#define SCALE    (1.0f/32768.0f)
#define MTILES   2500         // ceil(NFRAMES/16)
#define NTILES   17           // ceil(NBINS/16)
#define TWO_PI   6.28318530717958647f

typedef __attribute__((ext_vector_type(2))) float v2f;
typedef __attribute__((ext_vector_type(8))) float v8f;

// workspace layout (floats): [0]=sum(mic0) [1]=sum(mic1)
// [2 .. 2+4*257) = R00 | R11 | R01r | R01i accumulators
#define ACC_OFF 2
#define ACC_N   (4*NBINS)

// ---------------- K0: zero the accumulators -----------------
__global__ void k_zero(float* __restrict__ ws) {
  int i = blockIdx.x * blockDim.x + threadIdx.x;
  if (i < ACC_OFF + ACC_N) ws[i] = 0.0f;
}

// ---------------- K1: per-mic sums (for the mean) -----------------
__global__ void k_sums(const float* __restrict__ in0, const float* __restrict__ in1,
                       float* __restrict__ ws) {
  __shared__ float red[256];
  float s0 = 0.0f, s1 = 0.0f;
  for (int i = blockIdx.x * blockDim.x + threadIdx.x; i < T_SAMP;
       i += gridDim.x * blockDim.x) {
    s0 += in0[i];
    s1 += in1[i];
  }
  const int tid = threadIdx.x;
  red[tid] = s0; __syncthreads();
  for (int s = 128; s > 0; s >>= 1) { if (tid < s) red[tid] += red[tid + s]; __syncthreads(); }
  if (tid == 0) atomicAdd(&ws[0], red[0]);
  __syncthreads();
  red[tid] = s1; __syncthreads();
  for (int s = 128; s > 0; s >>= 1) { if (tid < s) red[tid] += red[tid + s]; __syncthreads(); }
  if (tid == 0) atomicAdd(&ws[1], red[0]);
}

// ---------------- K2: fused windowed-DFT (WMMA f32 16x16x4) + covariance ----
// One block = one 16-frame tile. 8 waves cover the 17 bin-tiles.
// LDS: two 16x512 f32 windowed/pre-emphasized frame tiles, XOR-swizzled.
__global__ void __launch_bounds__(256)
k_dft_wmma(const float* __restrict__ in0, const float* __restrict__ in1,
           float* __restrict__ ws, float invS) {
  extern __shared__ float smem[];
  float* y0 = smem;               // [16][512], index n XOR-swizzled by frame
  float* y1 = smem + 16 * NFFT;

  const int m0  = blockIdx.x * 16;     // first frame of this tile
  const int tid = threadIdx.x;
  const float mu0 = ws[0] * (SCALE / (float)T_SAMP);
  const float mu1 = ws[1] * (SCALE / (float)T_SAMP);

  // ---- stage: windowed, pre-emphasized, mean-subtracted frames ----
  for (int idx = tid; idx < 16 * NFFT; idx += 256) {
    const int fr = idx >> 9;
    const int n  = idx & (NFFT - 1);
    int t = (m0 + fr) * HOP + n;
    int tc = (t < T_SAMP) ? t : 0;               // clamp pad frames (weight=0 later)
    const float win = 0.5f - 0.5f * __cosf(TWO_PI * (float)n / (float)NFFT);
    const float a0 = in0[tc], a1 = in1[tc];
    const float p0 = in0[tc > 0 ? tc - 1 : 0], p1 = in1[tc > 0 ? tc - 1 : 0];
    float v0, v1;
    if (tc == 0) { v0 = SCALE * a0 - mu0; v1 = SCALE * a1 - mu1; }
    else {
      v0 = SCALE * (a0 - PRE * p0) - (1.0f - PRE) * mu0;
      v1 = SCALE * (a1 - PRE * p1) - (1.0f - PRE) * mu1;
    }
    const int sw = n ^ ((fr & 7) << 3);          // bank-conflict swizzle
    y0[fr * NFFT + sw] = win * v0;
    y1[fr * NFFT + sw] = win * v1;
  }
  __syncthreads();

  const int lane  = tid & 31;
  const int wv    = tid >> 5;
  const int mrow  = lane & 15;                   // A-matrix M row (ISA f32 A layout)
  const int off   = (lane >> 4) << 1;            // K sub-offset: 0 (lanes 0-15) / 2 (16-31)
  const int binl  = lane & 15;                   // B/C/D N index
  const int xmask = (mrow & 7) << 3;

  for (int nt = wv; nt < NTILES; nt += 8) {
    const int bin = nt * 16 + binl;
    // twiddle rotation state: (c,s) = (cos, sin)(2*pi*bin*n/512) for n = k0+off, k0+off+1
    const float phi = TWO_PI * (float)bin / (float)NFFT;
    float c0, s0v, c1, s1v, cr, sr;
    sincosf(phi * (float)off,       &s0v, &c0);
    sincosf(phi * (float)(off + 1), &s1v, &c1);
    sincosf(phi * 4.0f,             &sr,  &cr);  // per-k-step rotation (K advances by 4)

    v8f R0 = {}, I0 = {}, R1 = {}, I1 = {};
    const float* pa0 = y0 + mrow * NFFT;
    const float* pa1 = y1 + mrow * NFFT;

#pragma unroll 4
    for (int k0 = 0; k0 < NFFT; k0 += 4) {
      const int sidx = (k0 + off) ^ xmask;       // even -> 8B-aligned ds_load_b64
      v2f a0 = *(const v2f*)(pa0 + sidx);
      v2f a1 = *(const v2f*)(pa1 + sidx);
      v2f bc; bc.x = c0;   bc.y = c1;            // real basis:  cos
      v2f bs; bs.x = -s0v; bs.y = -s1v;          // imag basis: -sin
      R0 = __builtin_amdgcn_wmma_f32_16x16x4_f32(false, a0, false, bc, (short)0, R0, false, false);
      I0 = __builtin_amdgcn_wmma_f32_16x16x4_f32(false, a0, false, bs, (short)0, I0, false, false);
      R1 = __builtin_amdgcn_wmma_f32_16x16x4_f32(false, a1, false, bc, (short)0, R1, false, false);
      I1 = __builtin_amdgcn_wmma_f32_16x16x4_f32(false, a1, false, bs, (short)0, I1, false, false);
      // rotate twiddles by exp(-i*4*phi)
      float t0 = c0 * cr - s0v * sr; s0v = s0v * cr + c0 * sr; c0 = t0;
      float t1 = c1 * cr - s1v * sr; s1v = s1v * cr + c1 * sr; c1 = t1;
    }

    // ---- fused temporally-weighted covariance accumulation ----
    // C/D layout: element v of lane l -> M = v + 8*(l>>4), N = l&15
    float p00 = 0.0f, p11 = 0.0f, c01 = 0.0f, q01 = 0.0f;
    const int mhi = (lane >> 4) * 8;
#pragma unroll
    for (int v = 0; v < 8; ++v) {
      const int fg = m0 + v + mhi;
      const float w = (fg < NFRAMES)
          ? expf(-2.0f + 2.0f * (float)fg * (1.0f / 39999.0f)) * invS : 0.0f;
      const float r0 = R0[v], i0 = I0[v], r1 = R1[v], i1 = I1[v];
      p00 += w * (r0 * r0 + i0 * i0);
      p11 += w * (r1 * r1 + i1 * i1);
      c01 += w * (r0 * r1 + i0 * i1);
      q01 += w * (i0 * r1 - r0 * i1);
    }
    p00 += __shfl_xor(p00, 16);
    p11 += __shfl_xor(p11, 16);
    c01 += __shfl_xor(c01, 16);
    q01 += __shfl_xor(q01, 16);
    if (lane < 16 && bin < NBINS) {
      atomicAdd(&ws[ACC_OFF + 0 * NBINS + bin], p00);
      atomicAdd(&ws[ACC_OFF + 1 * NBINS + bin], p11);
      atomicAdd(&ws[ACC_OFF + 2 * NBINS + bin], c01);
      atomicAdd(&ws[ACC_OFF + 3 * NBINS + bin], q01);
    }
  }
}

// ---------------- K3: MVDR angle scan + argmax (tiny) -----------------
__global__ void k_score(const float* __restrict__ ws, float* __restrict__ out) {
  __shared__ float iR00[NBINS], iR11[NBINS], iR01r[NBINS], iR01i[NBINS], FW[NBINS];
  __shared__ float scores[NANGLES];
  __shared__ float sinv;
  const int tid = threadIdx.x;

  for (int k = tid; k < NBINS; k += blockDim.x) {
    const float R00 = ws[ACC_OFF + 0 * NBINS + k];
    const float R11 = ws[ACC_OFF + 1 * NBINS + k];
    const float R01r = ws[ACC_OFF + 2 * NBINS + k];
    const float R01i = ws[ACC_OFF + 3 * NBINS + k];
    const float trace = 0.5f * (R00 + R11);
    const float det = R00 * R11 - (R01r * R01r + R01i * R01i) + 1e-6f * trace;
    const float inv = 1.0f / det;
    iR00[k]  =  R11 * inv;
    iR11[k]  =  R00 * inv;
    iR01r[k] = -R01r * inv;
    iR01i[k] = -R01i * inv;
    // frequency weights
    const float f = (float)k * (16000.0f / 512.0f);
    const float srr = 0.05f * (f + 1e-6f) / 343.0f;
    const float spatial = 1.0f / (1.0f + expf(-4.0f * (srr - 0.1f)));
    const float d = f - 16000.0f / 6.0f;
    const float fwv = 1.0f - expf(-(d * d) / (2.0f * 4000.0f * 4000.0f));
    const float alias = (srr < 0.5f) ? 1.0f : expf(-2.0f * (srr - 0.5f) * (srr - 0.5f));
    FW[k] = spatial * fwv * alias;
  }
  __syncthreads();
  if (tid == 0) {
    float s = 0.0f;
    for (int k = 0; k < NBINS; ++k) s += FW[k];
    sinv = 1.0f / (s + 1e-6f);
  }
  __syncthreads();
  for (int k = tid; k < NBINS; k += blockDim.x) FW[k] *= sinv;
  __syncthreads();

  for (int a = tid; a < NANGLES; a += blockDim.x) {
    const float deg = (float)a;
    const float tau = -(0.05f * cosf(deg * 3.14159265358979323846f / 180.0f)) / 343.0f;
    const float delta = TWO_PI * tau * (16000.0f / 512.0f);  // phase step per bin
    float cd, sd; sincosf(delta, &sd, &cd);

    // pass 1: log-sum, arithmetic sum, softmax max
    float cr = 1.0f, sp = 0.0f;
    float slog = 0.0f, sar = 0.0f, mmax = -1e30f;
    for (int k = 0; k < NBINS; ++k) {
      const float qf = iR00[k] + iR11[k] + 2.0f * (iR01r[k] * cr + iR01i[k] * sp);
      const float mv = 1.0f / (qf + 1e-6f);
      slog += logf(mv + 1e-6f) * FW[k];
      sar  += mv * FW[k];
      mmax = fmaxf(mmax, mv + mv);
      const float nc = cr * cd - sp * sd; sp = sp * cd + cr * sd; cr = nc;
    }
    // pass 2: softmax-weighted peak
    cr = 1.0f; sp = 0.0f;
    float z = 0.0f, pk = 0.0f;
    for (int k = 0; k < NBINS; ++k) {
      const float qf = iR00[k] + iR11[k] + 2.0f * (iR01r[k] * cr + iR01i[k] * sp);
      const float mv = 1.0f / (qf + 1e-6f);
      const float e = expf(mv + mv - mmax);
      z  += e;
      pk += mv * e * FW[k];
      const float nc = cr * cd - sp * sd; sp = sp * cd + cr * sd; cr = nc;
    }
    scores[a] = 0.5f * expf(slog) + 0.3f * sar + 0.2f * (pk / z);
  }
  __syncthreads();
  if (tid == 0) {
    int best = 0; float bv = scores[0];
    for (int a = 1; a < NANGLES; ++a)
      if (scores[a] > bv) { bv = scores[a]; best = a; }
    out[0] = (float)best;   // ANGLE_GRID = 0..180 step 1
  }
}

// ---------------- launch -----------------
extern "C" void kernel_launch(void* const* d_in, const int* in_sizes, int n_in,
                              void* d_out, int out_size, void* d_ws, size_t ws_size,
                              hipStream_t stream) {
  const float* in0 = (const float*)d_in[0];
  const float* in1 = (const float*)d_in[1];
  float* ws  = (float*)d_ws;
  float* out = (float*)d_out;

  // TEMPORAL normalizer: sum over all 40000 weights (host scalar, deterministic)
  double S = 0.0;
  for (int i = 0; i < 40000; ++i) S += exp(-2.0 + 2.0 * (double)i / 39999.0);
  const float invS = (float)(1.0 / S);

  k_zero<<<(ACC_OFF + ACC_N + 255) / 256, 256, 0, stream>>>(ws);
  k_sums<<<1024, 256, 0, stream>>>(in0, in1, ws);
  const size_t shbytes = (size_t)2 * 16 * NFFT * sizeof(float);   // 64 KB
  k_dft_wmma<<<MTILES, 256, shbytes, stream>>>(in0, in1, ws, invS);
  k_score<<<1, 256, 0, stream>>>(ws, out);
}